// LabeledDividedLoss_22960895164560
// MI455X (gfx1250) — compile-verified
//
#include <hip/hip_runtime.h>
#include <hip/hip_bf16.h>
#include <math.h>

// ---------------------------------------------------------------------------
// LabeledDividedLoss on MI455X (gfx1250, wave32)
//  K1: one streaming pass over y1/y2 (512MB, NT b128 loads) -> per-row scalars
//  K2: WMMA f32 16x16x4 reduction of block partials (mean loss + KL total)
//  K3/K4: count loss<mean -> num_remember k
//  K5 x4: deterministic radix select of k-th smallest loss (uint keys, int atomics)
//  K6/K7/K8: exact tie-rank handling + masked sums (fixed-order reductions)
//  K9: final scalar
// ---------------------------------------------------------------------------

#define EPOCHS_F 200.0f
#define C_DIM    256
#define GRID1    2048   // must stay a multiple of 64 (K2 WMMA tiles)
#define BLK1     256

typedef float f32x2 __attribute__((ext_vector_type(2)));
typedef float f32x4 __attribute__((ext_vector_type(4)));
typedef float f32x8 __attribute__((ext_vector_type(8)));

struct Scal {
    float    mean_v;
    float    klTotal;
    int      k;
    unsigned prefix;     // becomes kth key after 4 radix rounds
    int      rank;       // remaining 0-indexed rank within current prefix
    int      countLess;  // #keys strictly < kth
    int      rememberTies;
    int      pad;
};

__device__ __forceinline__ float wredAdd(float v) {
#pragma unroll
    for (int m = 16; m > 0; m >>= 1) v += __shfl_xor(v, m, 32);
    return v;
}
__device__ __forceinline__ float wredMax(float v) {
#pragma unroll
    for (int m = 16; m > 0; m >>= 1) v = fmaxf(v, __shfl_xor(v, m, 32));
    return v;
}
__device__ __forceinline__ int wredMinI(int v) {
#pragma unroll
    for (int m = 16; m > 0; m >>= 1) v = min(v, __shfl_xor(v, m, 32));
    return v;
}
// uniform-index select out of an 8-register array (stays in VGPRs)
__device__ __forceinline__ float sel8(const float* v, int s) {
    float x = v[0];
    if (s == 1) x = v[1];
    if (s == 2) x = v[2];
    if (s == 3) x = v[3];
    if (s == 4) x = v[4];
    if (s == 5) x = v[5];
    if (s == 6) x = v[6];
    if (s == 7) x = v[7];
    return x;
}
// order-preserving f32 -> u32 key (ascending)
__device__ __forceinline__ unsigned f2key(float f) {
    unsigned u = __float_as_uint(f);
    return (u & 0x80000000u) ? ~u : (u | 0x80000000u);
}

// ---------------------------------------------------------------------------
// K1: per-row softmax statistics. One wave == one row (32 lanes x 8 cols).
// ---------------------------------------------------------------------------
__global__ void k1_rows(const float* __restrict__ y1, const float* __restrict__ y2,
                        const int* __restrict__ t, const int* __restrict__ ep,
                        float* __restrict__ lossA, float* __restrict__ wceA,
                        float* __restrict__ condA,
                        float* __restrict__ pL, float* __restrict__ pK, int nrows)
{
    __shared__ float sL[BLK1 / 32], sK[BLK1 / 32];
    const int lane = threadIdx.x & 31;
    const int wib  = threadIdx.x >> 5;
    const int wpb  = blockDim.x >> 5;
    const int wg   = blockIdx.x * wpb + wib;
    const int nw   = gridDim.x * wpb;

    const int   epoch  = ep[0];
    const float invn   = 1.0f / (float)nrows;
    const float thresh = 1.0f - (1.0f - fminf(0.5f, invn)) * (float)epoch / EPOCHS_F;
    const float wexp   = 0.5f - 0.5f * (float)epoch / EPOCHS_F;

    float accL = 0.0f, accK = 0.0f;

    for (int row = wg; row < nrows; row += nw) {
        const f32x4* r1 = (const f32x4*)(y1 + (size_t)row * C_DIM) + lane * 2;
        const f32x4* r2 = (const f32x4*)(y2 + (size_t)row * C_DIM) + lane * 2;
        if (row + nw < nrows) {  // global_prefetch_b8 for next row
            __builtin_prefetch(y1 + (size_t)(row + nw) * C_DIM + lane * 8, 0, 0);
            __builtin_prefetch(y2 + (size_t)(row + nw) * C_DIM + lane * 8, 0, 0);
        }
        // streamed once -> non-temporal b128 loads
        f32x4 a0 = __builtin_nontemporal_load(r1);
        f32x4 a1 = __builtin_nontemporal_load(r1 + 1);
        f32x4 b0 = __builtin_nontemporal_load(r2);
        f32x4 b1 = __builtin_nontemporal_load(r2 + 1);
        float v1[8] = {a0.x, a0.y, a0.z, a0.w, a1.x, a1.y, a1.z, a1.w};
        float v2[8] = {b0.x, b0.y, b0.z, b0.w, b1.x, b1.y, b1.z, b1.w};
        const int tr = t[row];

        float m1 = v1[0], m2 = v2[0];
#pragma unroll
        for (int j = 1; j < 8; ++j) { m1 = fmaxf(m1, v1[j]); m2 = fmaxf(m2, v2[j]); }
        m1 = wredMax(m1);
        m2 = wredMax(m2);

        float e1[8], e2[8], s1 = 0.0f, s2 = 0.0f;
#pragma unroll
        for (int j = 0; j < 8; ++j) {
            e1[j] = __expf(v1[j] - m1); s1 += e1[j];
            e2[j] = __expf(v2[j] - m2); s2 += e2[j];
        }
        s1 = wredAdd(s1);
        s2 = wredAdd(s2);
        const float L1 = __logf(s1), L2 = __logf(s2);
        const float lse1 = m1 + L1, lse2 = m2 + L2;
        const float i1 = 1.0f / s1, i2 = 1.0f / s2;  // == pmax1, pmax2

        float q11 = 0, q12 = 0, q21 = 0, q22 = 0;
#pragma unroll
        for (int j = 0; j < 8; ++j) {
            const float p1 = e1[j] * i1, p2 = e2[j] * i2;
            const float lp1 = v1[j] - lse1, lp2 = v2[j] - lse2;
            q11 = fmaf(p1, lp1, q11); q12 = fmaf(p1, lp2, q12);
            q21 = fmaf(p2, lp1, q21); q22 = fmaf(p2, lp2, q22);
        }
        q11 = wredAdd(q11); q12 = wredAdd(q12);
        q21 = wredAdd(q21); q22 = wredAdd(q22);

        // first-occurrence argmax (matches jnp.argmax)
        int ia1 = 0x7FFFFFFF, ia2 = 0x7FFFFFFF;
#pragma unroll
        for (int j = 7; j >= 0; --j) {
            if (v1[j] == m1) ia1 = lane * 8 + j;
            if (v2[j] == m2) ia2 = lane * 8 + j;
        }
        const int pred1 = wredMinI(ia1);
        const int pred2 = wredMinI(ia2);

        const float y1t = __shfl(sel8(v1, tr & 7), tr >> 3, 32);
        const float y2t = __shfl(sel8(v2, tr & 7), tr >> 3, 32);
        const float y2p = __shfl(sel8(v2, pred1 & 7), pred1 >> 3, 32);

        const float lossP = (lse1 - y1t) + (lse2 - y2t);
        const float klR   = (q22 - q21) + (q11 - q12);
        const float pp    = i1 * i2;                   // pmax1*pmax2
        const float cedc  = (lse1 - m1) + (lse2 - y2p);
        const float wv    = powf(pp, wexp);
        const int   cond  = (pred1 != tr) && (pred1 == pred2) && (pp > thresh);

        if (lane == 0) {
            lossA[row] = lossP;
            wceA[row]  = wv * cedc;
            condA[row] = cond ? 1.0f : 0.0f;
            accL += lossP;
            accK += klR;
        }
    }
    if (lane == 0) { sL[wib] = accL; sK[wib] = accK; }
    __syncthreads();
    if (threadIdx.x == 0) {  // fixed-order -> deterministic
        float tl = 0, tk = 0;
        for (int w = 0; w < wpb; ++w) { tl += sL[w]; tk += sK[w]; }
        pL[blockIdx.x] = tl;
        pK[blockIdx.x] = tk;
    }
}

// ---------------------------------------------------------------------------
// K2: WMMA f32 16x16x4 reduction of block partials. D = A*ones + C accumulates
// exactly in f32. Wave0 -> loss total (mean_v); wave1 -> KL total.
// A layout (ISA 7.12.2): lanes0-15 rows 0-15 K={0,1}; lanes16-31 K={2,3}.
// cnt MUST be a multiple of 64 -> no tail guards, single b64 load per step.
// ---------------------------------------------------------------------------
__global__ void k2_wmma_reduce(const float* __restrict__ pL,
                               const float* __restrict__ pK,
                               Scal* sc, int cnt, int nrows)
{
    const int lane = threadIdx.x & 31;
    const int wave = threadIdx.x >> 5;
    // lane's K-pair start within each 64-wide tile: rows m=lane&15, K base 0/2
    const int pairOff = ((lane & 15) * 4 + (lane >> 4) * 2) >> 1;  // in f32x2 units
    const f32x2* src = (const f32x2*)((wave == 0) ? pL : pK) + pairOff;
    const int steps = cnt >> 6;  // 64 values per WMMA tile

    f32x2 bOnes; bOnes.x = 1.0f; bOnes.y = 1.0f;
    f32x8 c = {};
    for (int it = 0; it < steps; ++it) {
        const f32x2 a = src[it * 32];  // one global_load_b64 per lane
        c = __builtin_amdgcn_wmma_f32_16x16x4_f32(false, a, false, bOnes,
                                                  (short)0, c, false, false);
    }
    // C/D: VGPR v at lanes<16 holds row v, lanes>=16 row v+8; all cols equal.
    float s = c[0] + c[1] + c[2] + c[3] + c[4] + c[5] + c[6] + c[7];
    s += __shfl_xor(s, 16, 32);
    if (lane == 0) {
        if (wave == 0) sc->mean_v  = s / (float)nrows;
        else           sc->klTotal = s;
    }
}

// K3: count loss < mean_v (integer partials, deterministic)
__global__ void k3_count(const float* __restrict__ lossA, const Scal* sc,
                         int* __restrict__ cntP, int n)
{
    __shared__ int sw[8];
    const int i = blockIdx.x * blockDim.x + threadIdx.x;
    const float mv = sc->mean_v;
    const int pred = (i < n) && (lossA[i] < mv);
    const unsigned long long bal = __ballot(pred);
    const int lane = threadIdx.x & 31, wib = threadIdx.x >> 5;
    if (lane == 0) sw[wib] = __popcll(bal);
    __syncthreads();
    if (threadIdx.x == 0) {
        int tot = 0;
        for (int w = 0; w < (int)(blockDim.x >> 5); ++w) tot += sw[w];
        cntP[blockIdx.x] = tot;
    }
}

// K4: total count -> k = num_remember; init radix-select state + hist
__global__ void k4_setup(const int* __restrict__ cntP, int nblocks,
                         const int* __restrict__ ep, Scal* sc,
                         unsigned* __restrict__ hist, int n)
{
    if (threadIdx.x < 256) hist[threadIdx.x] = 0u;
    if (threadIdx.x == 0) {
        long long tot = 0;
        for (int b = 0; b < nblocks; ++b) tot += cntP[b];
        const float rrs = (float)tot / (float)n;
        const float rr  = fmaxf(1.0f - (0.5f / EPOCHS_F) * (float)ep[0], rrs);
        int k = (int)floorf(rr * (float)n);
        if (k < 0) k = 0;
        if (k > n) k = n;
        sc->k = k;
        sc->prefix = 0u;
        sc->rank = (k > 0) ? (k - 1) : 0;
        sc->countLess = 0;
    }
}

// K5a: histogram of byte `digit` among keys matching current prefix
__global__ void k5_hist(const float* __restrict__ lossA, const Scal* sc,
                        unsigned* __restrict__ hist, int n, int digit)
{
    __shared__ unsigned lh[256];
    if (threadIdx.x < 256) lh[threadIdx.x] = 0u;
    __syncthreads();
    const int i = blockIdx.x * blockDim.x + threadIdx.x;
    const unsigned pfx = sc->prefix;
    if (i < n) {
        const unsigned key = f2key(lossA[i]);
        const bool ok = (digit == 3) || (((key ^ pfx) >> ((digit + 1) * 8)) == 0u);
        if (ok) atomicAdd(&lh[(key >> (digit * 8)) & 0xFFu], 1u);
    }
    __syncthreads();
    if (threadIdx.x < 256 && lh[threadIdx.x]) atomicAdd(&hist[threadIdx.x], lh[threadIdx.x]);
}

// K5b: pick bucket containing remaining rank; zero hist for next round
__global__ void k5_scan(Scal* sc, unsigned* hist, int digit)
{
    if (threadIdx.x != 0) return;
    if (sc->k <= 0) {
        sc->prefix = 0u; sc->rank = 0;
        for (int b = 0; b < 256; ++b) hist[b] = 0u;
        return;
    }
    const unsigned r = (unsigned)sc->rank;
    unsigned cum = 0, before = 0;
    int chosen = -1;
    for (int b = 0; b < 256; ++b) {
        const unsigned cb = hist[b];
        hist[b] = 0u;
        if (chosen < 0 && r < cum + cb) { chosen = b; before = cum; }
        cum += cb;
    }
    if (chosen < 0) { chosen = 255; before = cum; }  // defensive
    sc->prefix |= ((unsigned)chosen) << (digit * 8);
    int nr = (int)r - (int)before;
    if (nr < 0) nr = 0;
    sc->rank = nr;
    sc->countLess += (int)before;
}

// K6: per-block count of keys == kth key
__global__ void k6_ties(const float* __restrict__ lossA, const Scal* sc,
                        int* __restrict__ tieP, int n)
{
    __shared__ int sw[8];
    const int i = blockIdx.x * blockDim.x + threadIdx.x;
    const unsigned kth = sc->prefix;
    const int pred = (i < n) && (f2key(lossA[i]) == kth);
    const unsigned long long bal = __ballot(pred);
    const int lane = threadIdx.x & 31, wib = threadIdx.x >> 5;
    if (lane == 0) sw[wib] = __popcll(bal);
    __syncthreads();
    if (threadIdx.x == 0) {
        int tot = 0;
        for (int w = 0; w < (int)(blockDim.x >> 5); ++w) tot += sw[w];
        tieP[blockIdx.x] = tot;
    }
}

// K7: exclusive scan of tie counts (contiguous chunks -> global index order)
__global__ void k7_scan_ties(const int* __restrict__ tieP, int nblocks,
                             int* __restrict__ tieOff, Scal* sc)
{
    if (threadIdx.x != 0) return;
    int run = 0;
    for (int b = 0; b < nblocks; ++b) { tieOff[b] = run; run += tieP[b]; }
    int rt = sc->k - sc->countLess;
    if (rt < 0) rt = 0;
    sc->rememberTies = rt;
}

// K8: masked sums with exact argsort-compatible tie handling
__global__ void k8_final(const float* __restrict__ lossA, const float* __restrict__ wceA,
                         const float* __restrict__ condA, const Scal* sc,
                         const int* __restrict__ tieOff, float* __restrict__ p3, int n)
{
    __shared__ int   waveTies[8];
    __shared__ float sred[BLK1 * 3];
    const int i = blockIdx.x * blockDim.x + threadIdx.x;
    const int lane = threadIdx.x & 31, wib = threadIdx.x >> 5;
    const unsigned kth = sc->prefix;
    const int rTies = sc->rememberTies;

    float lossi = 0, wcei = 0, condi = 0;
    unsigned key = 0xFFFFFFFFu;
    if (i < n) { lossi = lossA[i]; wcei = wceA[i]; condi = condA[i]; key = f2key(lossi); }
    const bool tie = (i < n) && (key == kth);

    const unsigned mask = (unsigned)__ballot(tie);
    const int lanePre = __popc(mask & ((1u << lane) - 1u));
    if (lane == 0) waveTies[wib] = __popc(mask);
    __syncthreads();
    int waveOff = 0;
    for (int w = 0; w < wib; ++w) waveOff += waveTies[w];
    const int rank = tieOff[blockIdx.x] + waveOff + lanePre;  // global tie rank

    const bool remember = (i < n) && ((key < kth) || (tie && (rank < rTies)));
    const bool notupd   = (i < n) && !remember && (i != 0);
    const bool dcb      = notupd && (condi != 0.0f);
    const bool remb     = notupd && (condi == 0.0f);

    sred[threadIdx.x * 3 + 0] = remember ? lossi : 0.0f;
    sred[threadIdx.x * 3 + 1] = dcb ? wcei : 0.0f;
    sred[threadIdx.x * 3 + 2] = remb ? lossi : 0.0f;
    __syncthreads();
    if (threadIdx.x == 0) {  // fixed order -> deterministic
        float s0 = 0, s1 = 0, s2 = 0;
        for (int tdx = 0; tdx < (int)blockDim.x; ++tdx) {
            s0 += sred[tdx * 3 + 0];
            s1 += sred[tdx * 3 + 1];
            s2 += sred[tdx * 3 + 2];
        }
        p3[blockIdx.x * 3 + 0] = s0;
        p3[blockIdx.x * 3 + 1] = s1;
        p3[blockIdx.x * 3 + 2] = s2;
    }
}

// K9: final scalar = loss_clean + loss_dc + loss1 + 0.1 * inter_view
__global__ void k9_out(const float* __restrict__ p3, int nblocks,
                       const Scal* sc, float* __restrict__ out, int n)
{
    if (threadIdx.x != 0 || blockIdx.x != 0) return;
    float s0 = 0, s1 = 0, s2 = 0;
    for (int b = 0; b < nblocks; ++b) {
        s0 += p3[b * 3 + 0];
        s1 += p3[b * 3 + 1];
        s2 += p3[b * 3 + 2];
    }
    out[0] = (s0 + s1 + s2 + 0.1f * sc->klTotal) / (float)n;
}

// ---------------------------------------------------------------------------
extern "C" void kernel_launch(void* const* d_in, const int* in_sizes, int n_in,
                              void* d_out, int out_size, void* d_ws, size_t ws_size,
                              hipStream_t stream)
{
    const float* y1 = (const float*)d_in[0];
    const float* y2 = (const float*)d_in[1];
    const int*   t  = (const int*)d_in[2];
    const int*   ep = (const int*)d_in[3];
    const int n = in_sizes[2];  // number of rows (N)

    // workspace carve-out
    float* lossA = (float*)d_ws;
    float* wceA  = lossA + n;
    float* condA = wceA + n;
    float* pL    = condA + n;        // GRID1 entries
    float* pK    = pL + GRID1;
    const int B2 = (n + 255) / 256;
    int* cntP    = (int*)(pK + GRID1);
    int* tieP    = cntP + B2;
    int* tieOff  = tieP + B2;
    unsigned* hist = (unsigned*)(tieOff + B2);
    Scal* sc     = (Scal*)(hist + 256);
    float* p3    = (float*)(sc + 1);  // B2*3 entries

    k1_rows<<<GRID1, BLK1, 0, stream>>>(y1, y2, t, ep, lossA, wceA, condA, pL, pK, n);
    k2_wmma_reduce<<<1, 64, 0, stream>>>(pL, pK, sc, GRID1, n);
    k3_count<<<B2, 256, 0, stream>>>(lossA, sc, cntP, n);
    k4_setup<<<1, 256, 0, stream>>>(cntP, B2, ep, sc, hist, n);
    for (int d = 3; d >= 0; --d) {
        k5_hist<<<B2, 256, 0, stream>>>(lossA, sc, hist, n, d);
        k5_scan<<<1, 32, 0, stream>>>(sc, hist, d);
    }
    k6_ties<<<B2, 256, 0, stream>>>(lossA, sc, tieP, n);
    k7_scan_ties<<<1, 32, 0, stream>>>(tieP, B2, tieOff, sc);
    k8_final<<<B2, 256, 0, stream>>>(lossA, wceA, condA, sc, tieOff, p3, n);
    k9_out<<<1, 32, 0, stream>>>(p3, B2, sc, (float*)d_out, n);
}